// QuantumLayer_54924041781637
// MI455X (gfx1250) — compile-verified
//
#include <hip/hip_runtime.h>
#include <hip/hip_bf16.h>

// ---------------------------------------------------------------------------
// QuantumLayer for MI455X (gfx1250, wave32)
//   stage 1: partial q_in = x @ W_in^T       (WMMA f32 16x16x4, 16-way K-split
//            -> 8192 waves so HBM read BW (128 MB) can actually be saturated;
//            partial tiles land in d_ws, summed inside the sim kernel)
//   stage 2: 8-qubit statevector sim, 1 wave/sample, state in VGPRs+shfl_xor
//   stage 3: out = expz @ W_out^T + b_out    (WMMA f32 16x16x4, bias folded
//            into the accumulator init; 128 MB store-bound, 16384 waves)
// ---------------------------------------------------------------------------

#define BATCH  8192
#define INF    4096
#define NQ     8
#define NLAY   3
#define KSPLIT 16              // K-range per wave = 4096/16 = 256 = 8 chunks
#define NCHUNK (INF / KSPLIT / 32)

typedef float v2f __attribute__((ext_vector_type(2)));
typedef float v8f __attribute__((ext_vector_type(8)));

static __device__ __forceinline__ v8f wmma_f32(v2f a, v2f b, v8f c) {
  // D = A(16x4) * B(4x16) + C(16x16), all fp32
  return __builtin_amdgcn_wmma_f32_16x16x4_f32(false, a, false, b, (short)0, c,
                                               false, false);
}

// ========================= Stage 1: K-split GEMM ===========================
// wave id = (m-tile 0..511) * 16 + (k-split 0..15).  Each wave: 8 chunks of
// K=32, double-buffered in registers (16 b64 loads batched, then 8 WMMAs).
__global__ __launch_bounds__(256) void gemm_in_kernel(
    const float* __restrict__ x, const float* __restrict__ W_in,
    float* __restrict__ partial) {
  const int lane = threadIdx.x & 31;
  const int wid  = blockIdx.x * (blockDim.x >> 5) + (threadIdx.x >> 5); // 0..8191
  const int tile = wid >> 4;                 // m-tile
  const int ks   = wid & (KSPLIT - 1);       // k-split
  const int m0    = tile * 16;
  const int kbase = ks * (INF / KSPLIT);
  const int row = lane & 15;                 // A row (M) == B column (N)
  const int hi  = lane >> 4;                 // K sub-pair selector
  const float bscale = (row < NQ) ? 1.0f : 0.0f;
  const float* __restrict__ xrow = x    + (size_t)(m0 + row) * INF + kbase;
  const float* __restrict__ wrow = W_in + (size_t)(row & (NQ - 1)) * INF + kbase;

  v2f A[2][8], B[2][8];
#pragma unroll
  for (int j = 0; j < 8; ++j) {              // preload chunk 0
    const int k = j * 4 + 2 * hi;
    A[0][j] = *(const v2f*)(xrow + k);
    B[0][j] = *(const v2f*)(wrow + k);
  }

  v8f acc = {};
#pragma unroll
  for (int c = 0; c < NCHUNK; ++c) {
    const int cur = c & 1, nxt = cur ^ 1;
    if (c + 1 < NCHUNK) {
      __builtin_prefetch(xrow + (c + 2) * 32, 0, 3);
#pragma unroll
      for (int j = 0; j < 8; ++j) {          // batch-issue next chunk's loads
        const int k = (c + 1) * 32 + j * 4 + 2 * hi;
        A[nxt][j] = *(const v2f*)(xrow + k);
        B[nxt][j] = *(const v2f*)(wrow + k);
      }
    }
#pragma unroll
    for (int j = 0; j < 8; ++j) {
      v2f b = B[cur][j] * bscale;
      acc = wmma_f32(A[cur][j], b, acc);
    }
  }

  // store partial 16x8 tile (no bias here; summed in qsim_kernel)
  if (row < NQ) {
    float* __restrict__ p = partial + ((size_t)ks * BATCH) * NQ;
#pragma unroll
    for (int v = 0; v < 8; ++v)
      p[(size_t)(m0 + v + 8 * hi) * NQ + row] = acc[v];
  }
}

// ========================= Stage 2: 8-qubit simulator =======================
// Amplitude index a[7:0]:  a[7:3] = lane (wires 0..4), a[2:0] = reg slot
// (wires 5..7).  One wave per sample; gate unitaries wave-uniform.

__device__ __forceinline__ void gate_lane(float sr[8], float si[8], int lane, int m,
    float u00r, float u00i, float u01r, float u01i,
    float u10r, float u10i, float u11r, float u11i) {
  const bool h = (lane & m) != 0;
  const float c0r = h ? u10r : u00r, c0i = h ? u10i : u00i;
  const float c1r = h ? u11r : u01r, c1i = h ? u11i : u01i;
#pragma unroll
  for (int s = 0; s < 8; ++s) {
    const float pr = __shfl_xor(sr[s], m, 32);
    const float pi = __shfl_xor(si[s], m, 32);
    const float v0r = h ? pr : sr[s], v0i = h ? pi : si[s];
    const float v1r = h ? sr[s] : pr, v1i = h ? si[s] : pi;
    sr[s] = c0r * v0r - c0i * v0i + c1r * v1r - c1i * v1i;
    si[s] = c0r * v0i + c0i * v0r + c1r * v1i + c1i * v1r;
  }
}

__device__ __forceinline__ void gate_slot(float sr[8], float si[8], int b,
    float u00r, float u00i, float u01r, float u01i,
    float u10r, float u10i, float u11r, float u11i) {
#pragma unroll
  for (int s = 0; s < 8; ++s) {
    if (s & b) continue;
    const int t = s | b;
    const float v0r = sr[s], v0i = si[s], v1r = sr[t], v1i = si[t];
    sr[s] = u00r * v0r - u00i * v0i + u01r * v1r - u01i * v1i;
    si[s] = u00r * v0i + u00i * v0r + u01r * v1i + u01i * v1r;
    sr[t] = u10r * v0r - u10i * v0i + u11r * v1r - u11i * v1i;
    si[t] = u10r * v0i + u10i * v0r + u11r * v1i + u11i * v1r;
  }
}

__device__ __forceinline__ void apply_gate(float sr[8], float si[8], int lane, int wire,
    float u00r, float u00i, float u01r, float u01i,
    float u10r, float u10i, float u11r, float u11i) {
  if (wire < 5)
    gate_lane(sr, si, lane, 1 << (4 - wire),
              u00r, u00i, u01r, u01i, u10r, u10i, u11r, u11i);
  else
    gate_slot(sr, si, 1 << (7 - wire),
              u00r, u00i, u01r, u01i, u10r, u10i, u11r, u11i);
}

__device__ __forceinline__ void cnot_ll(float sr[8], float si[8], int lane, int cm, int tm) {
  const bool c = (lane & cm) != 0;
#pragma unroll
  for (int s = 0; s < 8; ++s) {
    const float pr = __shfl_xor(sr[s], tm, 32);
    const float pi = __shfl_xor(si[s], tm, 32);
    sr[s] = c ? pr : sr[s];
    si[s] = c ? pi : si[s];
  }
}
__device__ __forceinline__ void cnot_ls(float sr[8], float si[8], int lane, int cm, int tb) {
  const bool c = (lane & cm) != 0;
#pragma unroll
  for (int s = 0; s < 8; ++s) {
    if (s & tb) continue;
    const int t = s | tb;
    const float ar = sr[s], ai = si[s], br = sr[t], bi = si[t];
    sr[s] = c ? br : ar;  si[s] = c ? bi : ai;
    sr[t] = c ? ar : br;  si[t] = c ? ai : bi;
  }
}
__device__ __forceinline__ void cnot_sl(float sr[8], float si[8], int cb, int tm) {
#pragma unroll
  for (int s = 0; s < 8; ++s)
    if (s & cb) {
      sr[s] = __shfl_xor(sr[s], tm, 32);
      si[s] = __shfl_xor(si[s], tm, 32);
    }
}
__device__ __forceinline__ void cnot_ss(float sr[8], float si[8], int cb, int tb) {
#pragma unroll
  for (int s = 0; s < 8; ++s)
    if ((s & cb) && !(s & tb)) {
      const int t = s | tb;
      float tmp;
      tmp = sr[s]; sr[s] = sr[t]; sr[t] = tmp;
      tmp = si[s]; si[s] = si[t]; si[t] = tmp;
    }
}

__global__ __launch_bounds__(256) void qsim_kernel(
    const float* __restrict__ partial, const float* __restrict__ b_in,
    const float* __restrict__ qw, float* __restrict__ expz) {
  const int lane = threadIdx.x & 31;
  const int samp = blockIdx.x * (blockDim.x >> 5) + (threadIdx.x >> 5); // 0..8191

  // Reduce the 16 K-split partials: lane q (mod 8) owns angle q.
  const int q = lane & (NQ - 1);
  float angsum = b_in[q];
#pragma unroll
  for (int k = 0; k < KSPLIT; ++k)
    angsum += partial[((size_t)k * BATCH + samp) * NQ + q];

  float sr[8], si[8];
#pragma unroll
  for (int s = 0; s < 8; ++s) { sr[s] = 0.0f; si[s] = 0.0f; }
  sr[0] = (lane == 0) ? 1.0f : 0.0f;          // |0...0>

  // AngleEmbedding: RX(theta_i); angle broadcast from lane i
#pragma unroll
  for (int i = 0; i < NQ; ++i) {
    const float th = 0.5f * __shfl(angsum, i, 32);
    const float c = __cosf(th), s = __sinf(th);
    apply_gate(sr, si, lane, i,
               c, 0.0f, 0.0f, -s,
               0.0f, -s, c, 0.0f);
  }

  // 3 x (Rot on each wire, then CNOT ring)
#pragma unroll
  for (int l = 0; l < NLAY; ++l) {
#pragma unroll
    for (int i = 0; i < NQ; ++i) {
      const float phi   = qw[((l * NQ) + i) * 3 + 0];
      const float theta = qw[((l * NQ) + i) * 3 + 1];
      const float omega = qw[((l * NQ) + i) * 3 + 2];
      const float ct = __cosf(0.5f * theta), st = __sinf(0.5f * theta);
      const float al = 0.5f * (phi + omega), be = 0.5f * (phi - omega);
      const float ca = __cosf(al), sa = __sinf(al);
      const float cb = __cosf(be), sb = __sinf(be);
      // Rot = RZ(omega) RY(theta) RZ(phi)
      apply_gate(sr, si, lane, i,
                 ct * ca, -ct * sa,   -st * cb, -st * sb,
                 st * cb, -st * sb,    ct * ca,  ct * sa);
    }
    // CNOT ring: wire0..4 -> lane masks 16,8,4,2,1; wire5..7 -> slot bits 4,2,1
    cnot_ll(sr, si, lane, 16, 8);   // (0,1)
    cnot_ll(sr, si, lane, 8, 4);    // (1,2)
    cnot_ll(sr, si, lane, 4, 2);    // (2,3)
    cnot_ll(sr, si, lane, 2, 1);    // (3,4)
    cnot_ls(sr, si, lane, 1, 4);    // (4,5)
    cnot_ss(sr, si, 4, 2);          // (5,6)
    cnot_ss(sr, si, 2, 1);          // (6,7)
    cnot_sl(sr, si, 1, 16);         // (7,0)
  }

  // <Z_i> expectation values
  float p[8], P = 0.0f;
#pragma unroll
  for (int s = 0; s < 8; ++s) { p[s] = sr[s] * sr[s] + si[s] * si[s]; P += p[s]; }
  float z[8];
  z[0] = (lane & 16) ? -P : P;
  z[1] = (lane & 8)  ? -P : P;
  z[2] = (lane & 4)  ? -P : P;
  z[3] = (lane & 2)  ? -P : P;
  z[4] = (lane & 1)  ? -P : P;
  z[5] = (p[0] + p[1] + p[2] + p[3]) - (p[4] + p[5] + p[6] + p[7]);
  z[6] = (p[0] + p[1] + p[4] + p[5]) - (p[2] + p[3] + p[6] + p[7]);
  z[7] = (p[0] + p[2] + p[4] + p[6]) - (p[1] + p[3] + p[5] + p[7]);
#pragma unroll
  for (int off = 16; off; off >>= 1)
#pragma unroll
    for (int w = 0; w < 8; ++w)
      z[w] += __shfl_xor(z[w], off, 32);
  if (lane == 0) {
#pragma unroll
    for (int w = 0; w < 8; ++w)
      expz[(size_t)samp * NQ + w] = z[w];
  }
}

// ========================= Stage 3: out = expz @ W_out^T + b_out ============
// One wave per (16-col tile x 8 row tiles); B held in regs across the 8
// M-tiles; bias pre-loaded into the accumulator (C column per lane is fixed).
__global__ __launch_bounds__(256) void gemm_out_kernel(
    const float* __restrict__ expz, const float* __restrict__ W_out,
    const float* __restrict__ b_out, float* __restrict__ out) {
  const int lane = threadIdx.x & 31;
  const int wid  = blockIdx.x * (blockDim.x >> 5) + (threadIdx.x >> 5); // 0..16383
  const int ntile = wid >> 6;          // 0..255 : 16-col tile
  const int mgrp  = wid & 63;          // 0..63  : group of 8 row tiles
  const int n0 = ntile * 16;
  const int nn = lane & 15;
  const int hi = lane >> 4;

  const float* __restrict__ wrow = W_out + (size_t)(n0 + nn) * NQ;
  const v2f b0 = *(const v2f*)(wrow + 2 * hi);       // K = 0..3
  const v2f b1 = *(const v2f*)(wrow + 4 + 2 * hi);   // K = 4..7
  const float bias = b_out[n0 + nn];

  for (int t = 0; t < 8; ++t) {
    const int m0 = (mgrp * 8 + t) * 16;
    const float* __restrict__ arow = expz + (size_t)(m0 + nn) * NQ;
    const v2f a0 = *(const v2f*)(arow + 2 * hi);
    const v2f a1 = *(const v2f*)(arow + 4 + 2 * hi);
    v8f acc = {bias, bias, bias, bias, bias, bias, bias, bias};
    acc = wmma_f32(a0, b0, acc);
    acc = wmma_f32(a1, b1, acc);
#pragma unroll
    for (int v = 0; v < 8; ++v)
      out[(size_t)(m0 + v + 8 * hi) * INF + n0 + nn] = acc[v];
  }
}

// ============================================================================
extern "C" void kernel_launch(void* const* d_in, const int* in_sizes, int n_in,
                              void* d_out, int out_size, void* d_ws, size_t ws_size,
                              hipStream_t stream) {
  const float* x     = (const float*)d_in[0];
  const float* W_in  = (const float*)d_in[1];
  const float* b_in  = (const float*)d_in[2];
  const float* qw    = (const float*)d_in[3];
  const float* W_out = (const float*)d_in[4];
  const float* b_out = (const float*)d_in[5];
  float* out = (float*)d_out;

  float* partial = (float*)d_ws;                          // [16][8192][8]
  float* ez = partial + (size_t)KSPLIT * BATCH * NQ;      // [8192][8]

  // 512 m-tiles x 16 k-splits = 8192 waves / 8 per block
  gemm_in_kernel<<<1024, 256, 0, stream>>>(x, W_in, partial);
  // 8192 samples, one wave each (also reduces the K-split partials)
  qsim_kernel<<<1024, 256, 0, stream>>>(partial, b_in, qw, ez);
  // 16384 wave tasks / 8 per block
  gemm_out_kernel<<<2048, 256, 0, stream>>>(ez, W_out, b_out, out);
}